// PGActorContinuous_30666066494006
// MI455X (gfx1250) — compile-verified
//
#include <hip/hip_runtime.h>
#include <math.h>

typedef __attribute__((ext_vector_type(2))) float v2f;
typedef __attribute__((ext_vector_type(8))) float v8f;

#define SEQ_T   128
#define NFEAT   512
#define HID     1024
#define KTOT    1536          // NFEAT + HID
#define NWG     64            // one WG per 16-column tile (1024/16)
#define WGSIZE  256           // 8 wave32
#define NWAVE   8
#define KCHUNK  192           // K rows per wave (8 * 192 = 1536)
#define TILE_F  (KCHUNK * 16) // floats per wave tile region

// ---------------------------------------------------------------- grid sync
__device__ __forceinline__ void grid_sync(int* bar) {
    __syncthreads();
    if (threadIdx.x == 0) {
        __threadfence();   // release: h stores -> L2
        int g = __hip_atomic_load(&bar[1], __ATOMIC_RELAXED, __HIP_MEMORY_SCOPE_AGENT);
        int v = __hip_atomic_fetch_add(&bar[0], 1, __ATOMIC_ACQ_REL, __HIP_MEMORY_SCOPE_AGENT);
        if (v == NWG - 1) {
            __hip_atomic_store(&bar[0], 0, __ATOMIC_RELAXED, __HIP_MEMORY_SCOPE_AGENT);
            __hip_atomic_fetch_add(&bar[1], 1, __ATOMIC_RELEASE, __HIP_MEMORY_SCOPE_AGENT);
        } else {
            while (__hip_atomic_load(&bar[1], __ATOMIC_RELAXED, __HIP_MEMORY_SCOPE_AGENT) == g) {
                __builtin_amdgcn_s_sleep(1);
            }
        }
        __threadfence();
    }
    __syncthreads();
    __threadfence();       // acquire in every wave before reading the new h
}

// ---------------------------------------------------------------- init
__global__ void rnn_init_kernel(const float* __restrict__ h0,
                                float* __restrict__ hbuf, int* __restrict__ bar) {
    int i = blockIdx.x * blockDim.x + threadIdx.x;
    if (i < HID) hbuf[i] = h0[i];
    if (i == 0) { bar[0] = 0; bar[1] = 0; }
}

// ---------------------------------------------------------------- scan
__launch_bounds__(WGSIZE)
__global__ void rnn_scan_kernel(const float* __restrict__ x,
                                const float* __restrict__ Ws,
                                const float* __restrict__ bs,
                                float* __restrict__ hbuf,   // 2 * HID floats
                                int*   __restrict__ bar,
                                float* __restrict__ out) {  // out[2..1025] = h_final
    __shared__ float zsh[KTOT];
    __shared__ float psh[NWAVE * 16];
    __shared__ float wtile[2][NWAVE][TILE_F];   // double-buffered W step-tile (2*96KB)

    const int tid   = threadIdx.x;
    const int lane  = tid & 31;
    const int wave  = tid >> 5;
    const int colbase = blockIdx.x * 16;
    const int col16 = lane & 15;
    const int kw    = wave * KCHUNK;
    const int khalf = (lane >> 4) << 1;          // 0 (lanes 0-15) / 2 (lanes 16-31)
    const bool m0   = (lane & 15) == 0;          // lanes 0 & 16 carry A row M=0
    const int rlane = lane >> 2;                 // 0..7 : row within 8-row group
    const int c4    = (lane & 3) << 2;           // 0,4,8,12 : col within row

    float* h_cur = hbuf;
    float* h_nxt = hbuf + HID;

    // ---- async-stage W tile for t = 0 into buffer 0 (24 x b128 per wave)
    {
        const float* W0 = Ws;
        #pragma unroll
        for (int i = 0; i < 24; ++i) {
            int lr = i * 8 + rlane;                               // 0..191
            const float* g = W0 + (size_t)(kw + lr) * HID + colbase + c4;
            unsigned l = (unsigned)(size_t)&wtile[0][wave][lr * 16 + c4];
            asm volatile("global_load_async_to_lds_b128 %0, %1, off"
                         :: "v"(l), "v"(g) : "memory");
        }
    }

    for (int t = 0; t < SEQ_T; ++t) {
        const int buf = t & 1;

        // ---- issue async stage of step t+1 into the other buffer, then wait
        //      for step t's tile (async loads complete in order => cnt<=24 ok)
        if (t + 1 < SEQ_T) {
            const float* W1 = Ws + (size_t)(t + 1) * ((size_t)KTOT * HID);
            #pragma unroll
            for (int i = 0; i < 24; ++i) {
                int lr = i * 8 + rlane;
                const float* g = W1 + (size_t)(kw + lr) * HID + colbase + c4;
                unsigned l = (unsigned)(size_t)&wtile[buf ^ 1][wave][lr * 16 + c4];
                asm volatile("global_load_async_to_lds_b128 %0, %1, off"
                             :: "v"(l), "v"(g) : "memory");
            }
            asm volatile("s_wait_asynccnt 0x18" ::: "memory");   // <=24 outstanding
        } else {
            asm volatile("s_wait_asynccnt 0x0" ::: "memory");
        }

        // ---- stage z = [x_t | h_cur] into LDS (uniform 6 iters/thread)
        for (int i = tid; i < KTOT; i += WGSIZE)
            zsh[i] = (i < NFEAT) ? x[t * NFEAT + i] : h_cur[i - NFEAT];
        __syncthreads();

        // ---- 48 chained WMMAs fed from LDS
        const float* wl = &wtile[buf][wave][0];
        v8f c = {};
        #pragma unroll
        for (int j = 0; j < KCHUNK / 4; ++j) {
            int r = 4 * j + khalf;
            float za = zsh[kw + r];
            float zb = zsh[kw + r + 1];
            v2f a, b;
            a[0] = m0 ? za : 0.0f;
            a[1] = m0 ? zb : 0.0f;
            b[0] = wl[r * 16 + col16];              // row r      (K, K+2)
            b[1] = wl[(r + 1) * 16 + col16];        // row r+1    (K+1, K+3)
            c = __builtin_amdgcn_wmma_f32_16x16x4_f32(
                    false, a, false, b, (short)0, c, false, false);
        }

        // ---- C row 0 (VGPR0, lanes 0..15) -> LDS partials, reduce, tanh
        if (lane < 16) psh[wave * 16 + lane] = c[0];
        __syncthreads();

        if (tid < 16) {
            float s = bs[t * HID + colbase + tid];
            #pragma unroll
            for (int w2 = 0; w2 < NWAVE; ++w2) s += psh[w2 * 16 + tid];
            float hn = tanhf(s);
            h_nxt[colbase + tid] = hn;
            if (t == SEQ_T - 1) out[2 + colbase + tid] = hn;
        }

        grid_sync(bar);
        float* tmp = h_cur; h_cur = h_nxt; h_nxt = tmp;
    }
}

// ---------------------------------------------------------------- head
__global__ void rnn_head_kernel(const float* __restrict__ Wout,
                                const float* __restrict__ bout,
                                float* __restrict__ out) {
    __shared__ float r0[256], r1[256];
    int tid = threadIdx.x;
    float s0 = 0.f, s1 = 0.f;
    for (int i = tid; i < HID; i += 256) {
        float hv = out[2 + i];
        s0 += hv * Wout[2 * i + 0];
        s1 += hv * Wout[2 * i + 1];
    }
    r0[tid] = s0; r1[tid] = s1;
    __syncthreads();
    for (int off = 128; off > 0; off >>= 1) {
        if (tid < off) { r0[tid] += r0[tid + off]; r1[tid] += r1[tid + off]; }
        __syncthreads();
    }
    if (tid == 0) {
        out[0] = r0[0] + bout[0];
        out[1] = r1[0] + bout[1];
    }
}

// ---------------------------------------------------------------- launch
extern "C" void kernel_launch(void* const* d_in, const int* in_sizes, int n_in,
                              void* d_out, int out_size, void* d_ws, size_t ws_size,
                              hipStream_t stream) {
    const float* x    = (const float*)d_in[0];   // (1,128,512)
    const float* h0   = (const float*)d_in[1];   // (1024,)
    const float* Ws   = (const float*)d_in[2];   // (128,1536,1024)
    const float* bs   = (const float*)d_in[3];   // (128,1024)
    const float* Wout = (const float*)d_in[4];   // (1024,2)
    const float* bout = (const float*)d_in[5];   // (2,)
    float* out = (float*)d_out;                  // [actors(2) | h_final(1024)]

    float* hbuf = (float*)d_ws;                          // 2*HID floats
    int*   bar  = (int*)((char*)d_ws + 2 * HID * sizeof(float));

    rnn_init_kernel<<<4, 256, 0, stream>>>(h0, hbuf, bar);
    rnn_scan_kernel<<<NWG, WGSIZE, 0, stream>>>(x, Ws, bs, hbuf, bar, out);
    rnn_head_kernel<<<1, 256, 0, stream>>>(Wout, bout, out);
}